// RANCoders_37340445671918
// MI455X (gfx1250) — compile-verified
//
#include <hip/hip_runtime.h>

// ---------------------------------------------------------------------------
// RANCoders: E=100 estimators, all layers linear => collapse each estimator to
//   o_hi[e] = xg[e] @ M_hi[e] + c_hi[e],   M_hi[e]: [10,100], c_hi[e]: [100]
// Stage 1: compose M/c per estimator (tiny), storing M in WMMA-B-fragment
// order so stage 2 loads each v2f B fragment with a single b64 load.
// Stage 2: V_WMMA_F32_16X16X4_F32 GEMM; store-bandwidth bound
// (655 MB of fp32 output => ~28us floor at 23.3 TB/s).
// ---------------------------------------------------------------------------

typedef float v2f __attribute__((ext_vector_type(2)));
typedef float v8f __attribute__((ext_vector_type(8)));

#define NEST 100   // estimators
#define NFEA 10    // features per estimator
#define DIN  100   // input dim
#define NBAT 8192  // batch
#define KP   12    // K padded (10 -> 12, three K=4 WMMA steps)
#define DP   112   // D padded (100 -> 7*16)

// Packed M layout per estimator (floats): [chunk 0..2][hf 0..1][d 0..111][pair 0..1]
// value = M[k][d] with k = chunk*4 + hf*2 + pair (zero-padded k>=10, d>=100).
// Size per estimator: 3*2*112*2 = 1344 floats (same as KP*DP).

// workspace layout (in floats)
#define MHI_OFF 0
#define MLO_OFF (NEST * KP * DP)                    // 134400
#define CHI_OFF (2 * NEST * KP * DP)                // 268800
#define CLO_OFF (2 * NEST * KP * DP + NEST * DP)    // 280000

// ---------------------------------------------------------------------------
// Stage 1: per-estimator composition of the collapsed affine map.
// One 128-thread block per estimator. ~30K MACs each — negligible.
// ---------------------------------------------------------------------------
__global__ __launch_bounds__(128) void compose_kernel(
    const float* __restrict__ ew0, const float* __restrict__ eb0,
    const float* __restrict__ ew1, const float* __restrict__ eb1,
    const float* __restrict__ ew2, const float* __restrict__ eb2,
    const float* __restrict__ hw0, const float* __restrict__ hb0,
    const float* __restrict__ hw1, const float* __restrict__ hb1,
    const float* __restrict__ hw2, const float* __restrict__ hb2,
    const float* __restrict__ lw0, const float* __restrict__ lb0,
    const float* __restrict__ lw1, const float* __restrict__ lb1,
    const float* __restrict__ lw2, const float* __restrict__ lb2,
    float* __restrict__ ws)
{
  const int e   = blockIdx.x;
  const int tid = threadIdx.x;

  __shared__ float sT1[10][2];    // enc_w0 @ enc_w1
  __shared__ float sMenc[10][2];  // 10x2 collapsed encoder
  __shared__ float sbenc[2];
  __shared__ float sU[2][50];     // dec_w0 @ dec_w1
  __shared__ float sbu[50];
  __shared__ float sWd[2][100];   // 2x100 collapsed decoder
  __shared__ float sbd[100];

  const float* w0 = ew0 + e * 10 * 5;
  const float* w1 = ew1 + e * 5 * 2;
  const float* w2 = ew2 + e * 2 * 2;

  if (tid < 20) {
    int i = tid >> 1, k = tid & 1;
    float s = 0.f;
    for (int l = 0; l < 5; ++l) s += w0[i * 5 + l] * w1[l * 2 + k];
    sT1[i][k] = s;
  }
  __syncthreads();
  if (tid < 20) {
    int i = tid >> 1, j = tid & 1;
    sMenc[i][j] = sT1[i][0] * w2[0 * 2 + j] + sT1[i][1] * w2[1 * 2 + j];
  }
  if (tid < 2) {
    int j = tid;
    const float* b0 = eb0 + e * 5;
    const float* b1 = eb1 + e * 2;
    const float* b2 = eb2 + e * 2;
    float tb0 = b1[0], tb1 = b1[1];
    for (int l = 0; l < 5; ++l) {
      tb0 += b0[l] * w1[l * 2 + 0];
      tb1 += b0[l] * w1[l * 2 + 1];
    }
    sbenc[j] = tb0 * w2[0 * 2 + j] + tb1 * w2[1 * 2 + j] + b2[j];
  }
  __syncthreads();

  for (int which = 0; which < 2; ++which) {
    const float* dw0 = (which == 0 ? hw0 : lw0) + e * 2 * 25;
    const float* dw1 = (which == 0 ? hw1 : lw1) + e * 25 * 50;
    const float* dw2 = (which == 0 ? hw2 : lw2) + e * 50 * 100;
    const float* db0 = (which == 0 ? hb0 : lb0) + e * 25;
    const float* db1 = (which == 0 ? hb1 : lb1) + e * 50;
    const float* db2 = (which == 0 ? hb2 : lb2) + e * 100;

    if (tid < 100) {            // U = dw0(2x25) @ dw1(25x50)
      int a = tid / 50, c = tid % 50;
      float s = 0.f;
      for (int b = 0; b < 25; ++b) s += dw0[a * 25 + b] * dw1[b * 50 + c];
      sU[a][c] = s;
    }
    if (tid < 50) {             // bu = db0 @ dw1 + db1
      float s = db1[tid];
      for (int b = 0; b < 25; ++b) s += db0[b] * dw1[b * 50 + tid];
      sbu[tid] = s;
    }
    __syncthreads();
    for (int idx = tid; idx < 200; idx += 128) {  // Wd = U @ dw2(50x100)
      int a = idx / 100, d = idx % 100;
      float s = 0.f;
      for (int c = 0; c < 50; ++c) s += sU[a][c] * dw2[c * 100 + d];
      sWd[a][d] = s;
    }
    if (tid < 100) {            // bd = bu @ dw2 + db2
      float s = db2[tid];
      for (int c = 0; c < 50; ++c) s += sbu[c] * dw2[c * 100 + tid];
      sbd[tid] = s;
    }
    __syncthreads();

    // M[e] = Menc(10x2) @ Wd(2x100), written in packed B-fragment order:
    // idx = ((chunk*2 + hf)*DP + d)*2 + pair,  k = chunk*4 + hf*2 + pair
    float* Mo = ws + (which == 0 ? MHI_OFF : MLO_OFF) + e * KP * DP;
    float* co = ws + (which == 0 ? CHI_OFF : CLO_OFF) + e * DP;
    for (int idx = tid; idx < KP * DP; idx += 128) {
      int p  = idx & 1;
      int d  = (idx >> 1) % DP;
      int ch = (idx >> 1) / DP;        // 0..5 = chunk*2 + hf
      int k  = (ch >> 1) * 4 + (ch & 1) * 2 + p;
      float v = 0.f;
      if (k < 10 && d < 100) v = sMenc[k][0] * sWd[0][d] + sMenc[k][1] * sWd[1][d];
      Mo[idx] = v;
    }
    for (int idx = tid; idx < DP; idx += 128) {
      float v = 0.f;
      if (idx < 100) v = sbenc[0] * sWd[0][idx] + sbenc[1] * sWd[1][idx] + sbd[idx];
      co[idx] = v;
    }
    __syncthreads();  // protect shared reuse for the 'lo' pass
  }
}

// ---------------------------------------------------------------------------
// Stage 2: per-wave 16x16 output tiles via V_WMMA_F32_16X16X4_F32.
// A (16x12 gathered features) loaded once per wave, reused across 7 col tiles
// x {hi, lo} => 42 WMMAs per wave. B fragments are single b64 loads thanks to
// the packed layout; accumulators start from inline-0 SRC2 (bias added at
// store). Wave-uniform exits only, so EXEC is all-ones at every WMMA.
// ---------------------------------------------------------------------------
__global__ __launch_bounds__(256) void ens_wmma_kernel(
    const float* __restrict__ inputs, const int* __restrict__ rs,
    const float* __restrict__ ws, float* __restrict__ out)
{
  const float* chi = ws + CHI_OFF;
  const float* clo = ws + CLO_OFF;
  float* ohi = out;
  float* olo = out + (size_t)NEST * NBAT * DIN;

  const int lane = threadIdx.x & 31;
  const int wid  = (int)((blockIdx.x * blockDim.x + threadIdx.x) >> 5);
  if (wid >= NEST * (NBAT / 16)) return;  // wave-uniform

  const int e  = wid >> 9;        // 512 = NBAT/16 tiles per estimator
  const int t  = wid & 511;
  const int n0 = t << 4;
  const int row = lane & 15;
  const int hf  = lane >> 4;      // half-wave selector: K pair {0,1} vs {2,3}

  // Gather A fragment: 16 rows x K (10 features, padded to 12)
  const float* xrow = inputs + (size_t)(n0 + row) * DIN;
  const int*   rse  = rs + e * NFEA;
  const int kb = 2 * hf;
  v2f a0, a1, a2;
  a0.x = xrow[rse[kb + 0]];  a0.y = xrow[rse[kb + 1]];   // K 0..3
  a1.x = xrow[rse[kb + 4]];  a1.y = xrow[rse[kb + 5]];   // K 4..7
  a2.x = hf ? 0.f : xrow[rse[8]];                        // K 8..11 (10,11 pad)
  a2.y = hf ? 0.f : xrow[rse[9]];

  // Packed B-fragment views: entry (ch, d) is the v2f for K pair of chunk/hf.
  const v2f* Mh = (const v2f*)(ws + MHI_OFF + e * KP * DP);  // 6*DP v2f entries
  const v2f* Ml = (const v2f*)(ws + MLO_OFF + e * KP * DP);
  const float* ch = chi + e * DP;
  const float* cl = clo + e * DP;
  const size_t obase = ((size_t)e * NBAT + n0 + 8 * hf) * DIN;

#pragma unroll
  for (int tt = 0; tt < 7; ++tt) {
    const int d = tt * 16 + row;  // output column for this lane (padded space)

    // ---- hi ----
    {
      const v2f b0 = Mh[(0 + hf) * DP + d];
      const v2f b1 = Mh[(2 + hf) * DP + d];
      const v2f b2 = Mh[(4 + hf) * DP + d];
      v8f acc = {};  // inline 0 C-matrix
      acc = __builtin_amdgcn_wmma_f32_16x16x4_f32(false, a0, false, b0, (short)0, acc, false, false);
      acc = __builtin_amdgcn_wmma_f32_16x16x4_f32(false, a1, false, b1, (short)0, acc, false, false);
      acc = __builtin_amdgcn_wmma_f32_16x16x4_f32(false, a2, false, b2, (short)0, acc, false, false);
      if (d < DIN) {
        const float bias = ch[d];
        float* po = ohi + obase + d;
#pragma unroll
        for (int v = 0; v < 8; ++v) po[(size_t)v * DIN] = acc[v] + bias;
      }
    }

    // ---- lo ----
    {
      const v2f b0 = Ml[(0 + hf) * DP + d];
      const v2f b1 = Ml[(2 + hf) * DP + d];
      const v2f b2 = Ml[(4 + hf) * DP + d];
      v8f acc = {};
      acc = __builtin_amdgcn_wmma_f32_16x16x4_f32(false, a0, false, b0, (short)0, acc, false, false);
      acc = __builtin_amdgcn_wmma_f32_16x16x4_f32(false, a1, false, b1, (short)0, acc, false, false);
      acc = __builtin_amdgcn_wmma_f32_16x16x4_f32(false, a2, false, b2, (short)0, acc, false, false);
      if (d < DIN) {
        const float bias = cl[d];
        float* po = olo + obase + d;
#pragma unroll
        for (int v = 0; v < 8; ++v) po[(size_t)v * DIN] = acc[v] + bias;
      }
    }
  }
}

// ---------------------------------------------------------------------------
extern "C" void kernel_launch(void* const* d_in, const int* in_sizes, int n_in,
                              void* d_out, int out_size, void* d_ws, size_t ws_size,
                              hipStream_t stream) {
  const float* inputs = (const float*)d_in[0];
  const int*   rsamp  = (const int*)d_in[1];
  const float* ew0 = (const float*)d_in[2],  *eb0 = (const float*)d_in[3];
  const float* hw0 = (const float*)d_in[4],  *hb0 = (const float*)d_in[5];
  const float* lw0 = (const float*)d_in[6],  *lb0 = (const float*)d_in[7];
  const float* ew1 = (const float*)d_in[8],  *eb1 = (const float*)d_in[9];
  const float* hw1 = (const float*)d_in[10], *hb1 = (const float*)d_in[11];
  const float* lw1 = (const float*)d_in[12], *lb1 = (const float*)d_in[13];
  const float* ew2 = (const float*)d_in[14], *eb2 = (const float*)d_in[15];
  const float* hw2 = (const float*)d_in[16], *hb2 = (const float*)d_in[17];
  const float* lw2 = (const float*)d_in[18], *lb2 = (const float*)d_in[19];
  float* ws = (float*)d_ws;

  compose_kernel<<<NEST, 128, 0, stream>>>(ew0, eb0, ew1, eb1, ew2, eb2,
                                           hw0, hb0, hw1, hb1, hw2, hb2,
                                           lw0, lb0, lw1, lb1, lw2, lb2, ws);

  const int totalWaves = NEST * (NBAT / 16);          // 51200
  const int blocks = (totalWaves * 32 + 255) / 256;   // 6400
  ens_wmma_kernel<<<blocks, 256, 0, stream>>>(inputs, rsamp, ws, (float*)d_out);
}